// PostProcessor_75419625717922
// MI455X (gfx1250) — compile-verified
//
#include <hip/hip_runtime.h>

#define NPROP   2048
#define NCLS    81
#define NFG     80
#define IMG_W   1333.0f
#define IMG_H   800.0f
#define SCORE_T 0.05f
#define NMS_T   0.5f
#define NDET    100
#define BBOX_CLIP 4.135166556742356f   // log(1000/16)

// ---------------- CDNA5 async global->LDS staging (gfx1250) ----------------
#if defined(__AMDGCN__) && __has_builtin(__builtin_amdgcn_global_load_async_to_lds_b128)
#define USE_ASYNC_LDS 1
#else
#define USE_ASYNC_LDS 0
#endif

#if USE_ASYNC_LDS
typedef int v4i_t __attribute__((vector_size(16)));
__device__ __forceinline__ void async_g2l_16(const void* g, void* l) {
  // (global src, LDS dst, imm offset, imm cpol) - generic pointers, codegen casts
  __builtin_amdgcn_global_load_async_to_lds_b128((v4i_t*)g, (v4i_t*)l, 0, 0);
}
__device__ __forceinline__ void wait_async0() {
#if __has_builtin(__builtin_amdgcn_s_wait_asynccnt)
  __builtin_amdgcn_s_wait_asynccnt(0);
#else
  asm volatile("s_wait_asynccnt 0" ::: "memory");
#endif
}
#endif

// ---------------- monotone sort keys: (score desc, index asc) -------------
__device__ __forceinline__ unsigned long long enc_key(float s, unsigned idx) {
  unsigned u = __float_as_uint(s);
  unsigned ou = (u & 0x80000000u) ? ~u : (u | 0x80000000u);  // order-preserving
  return ((unsigned long long)ou << 32) | (unsigned)(~idx);  // bigger = better
}
__device__ __forceinline__ float dec_score(unsigned ou) {
  unsigned u = (ou & 0x80000000u) ? (ou & 0x7FFFFFFFu) : ~ou;
  return __uint_as_float(u);
}

// =========== Kernel A: softmax + decode + clip (1 wave per proposal) ======
__global__ __launch_bounds__(256) void kA(const float* __restrict__ logits,
                                          const float* __restrict__ reg,
                                          const float* __restrict__ props,
                                          float* __restrict__ scores_c,
                                          float* __restrict__ boxes_c) {
  __shared__ float sreg[8 * 324];            // 8 proposals x 81 classes x 4
  const int tid  = threadIdx.x;
  const int wave = tid >> 5, lane = tid & 31;
  const int pbase = blockIdx.x * 8;
  const float* gsrc = reg + (size_t)pbase * 324;

  __builtin_prefetch(logits + (size_t)pbase * NCLS, 0, 3);  // -> global_prefetch_b8

#if USE_ASYNC_LDS
  for (int c = tid; c < 648; c += 256)       // 648 x 16B = 10368B tile
    async_g2l_16(gsrc + c * 4, &sreg[c * 4]);
#else
  for (int t = tid; t < 8 * 324; t += 256) sreg[t] = gsrc[t];
#endif

  // ---- softmax over 81 classes, one wave32 per row (overlaps the DMA) ----
  const int p = pbase + wave;
  const float* lrow = logits + (size_t)p * NCLS;
  float v0 = lrow[lane];
  float v1 = lrow[lane + 32];
  float v2 = (lane < 17) ? lrow[lane + 64] : -3.0e38f;
  float m = fmaxf(v0, fmaxf(v1, v2));
#pragma unroll
  for (int o = 16; o; o >>= 1) m = fmaxf(m, __shfl_xor(m, o, 32));
  float e0 = expf(v0 - m), e1 = expf(v1 - m);
  float e2 = (lane < 17) ? expf(v2 - m) : 0.0f;
  float s = e0 + e1 + e2;
#pragma unroll
  for (int o = 16; o; o >>= 1) s += __shfl_xor(s, o, 32);
  float inv = 1.0f / s;
  // drop background class 0; layout scores_c[class-1][p]
  if (lane >= 1) scores_c[(lane - 1) * NPROP + p] = e0 * inv;
  scores_c[(lane + 31) * NPROP + p] = e1 * inv;
  if (lane < 17) scores_c[(lane + 63) * NPROP + p] = e2 * inv;

#if USE_ASYNC_LDS
  wait_async0();
#endif
  __syncthreads();

  // ---- decode + clip, classes 1..80 striped over 32 lanes ----
  float px1 = props[p * 4 + 0], py1 = props[p * 4 + 1];
  float px2 = props[p * 4 + 2], py2 = props[p * 4 + 3];
  float w = px2 - px1 + 1.0f, h = py2 - py1 + 1.0f;
  float cx = px1 + 0.5f * w,  cy = py1 + 0.5f * h;
  const float* rrow = &sreg[wave * 324];
  for (int ci = lane; ci < NFG; ci += 32) {      // ci = class-1
    const float* r = rrow + (ci + 1) * 4;
    float dx = r[0] / 10.0f, dy = r[1] / 10.0f;
    float dw = fminf(r[2] / 5.0f, BBOX_CLIP);
    float dh = fminf(r[3] / 5.0f, BBOX_CLIP);
    float pcx = dx * w + cx, pcy = dy * h + cy;
    float pw = expf(dw) * w, ph = expf(dh) * h;
    float x1 = pcx - 0.5f * pw, y1 = pcy - 0.5f * ph;
    float x2 = pcx + 0.5f * pw - 1.0f, y2 = pcy + 0.5f * ph - 1.0f;
    x1 = fminf(fmaxf(x1, 0.0f), IMG_W - 1.0f);
    y1 = fminf(fmaxf(y1, 0.0f), IMG_H - 1.0f);
    x2 = fminf(fmaxf(x2, 0.0f), IMG_W - 1.0f);
    y2 = fminf(fmaxf(y2, 0.0f), IMG_H - 1.0f);
    float4 b; b.x = x1; b.y = y1; b.z = x2; b.w = y2;
    ((float4*)boxes_c)[ci * NPROP + p] = b;
  }
}

// =========== Kernel B: per-class NMS (1 block per class) ==================
__global__ __launch_bounds__(256) void kB(const float* __restrict__ scores_c,
                                          const float* __restrict__ boxes_c,
                                          float* __restrict__ masked) {
  __shared__ unsigned long long K[NPROP];   // 16 KB
  __shared__ float4 BX[NPROP];              // 32 KB
  __shared__ unsigned char SUP[NPROP];      // 2 KB
  __shared__ int scnt;
  const int tid = threadIdx.x;
  const int c = blockIdx.x;
  if (tid == 0) scnt = 0;
  __syncthreads();

  int local = 0;
  for (int t = tid; t < NPROP; t += 256) {
    float s = scores_c[c * NPROP + t];
    bool v = s > SCORE_T;
    K[t] = v ? enc_key(s, (unsigned)t) : 0ull;
    masked[c * NPROP + t] = -1.0f;
    if (v) local++;
  }
  atomicAdd(&scnt, local);
  __syncthreads();

  // bitonic sort, descending (score desc, index asc; invalids=0 sink to end)
  for (int kk = 2; kk <= NPROP; kk <<= 1) {
    for (int j = kk >> 1; j > 0; j >>= 1) {
      for (int t = tid; t < NPROP; t += 256) {
        int ixj = t ^ j;
        if (ixj > t) {
          unsigned long long a = K[t], b = K[ixj];
          bool up = (t & kk) == 0;
          if (up ? (a < b) : (a > b)) { K[t] = b; K[ixj] = a; }
        }
      }
      __syncthreads();
    }
  }
  const int cnt = scnt;
  for (int i = tid; i < cnt; i += 256) {
    unsigned id = ~(unsigned)K[i];
    BX[i] = ((const float4*)boxes_c)[c * NPROP + id];
    SUP[i] = 0;
  }
  __syncthreads();

  // greedy NMS on the compacted, sorted candidates (cnt ~ 60 typical)
  for (int i = 0; i < cnt; ++i) {
    if (!SUP[i]) {                           // uniform read after barrier
      unsigned long long ki = K[i];
      float4 bi = BX[i];
      if (tid == 0) {
        unsigned id = ~(unsigned)ki;
        masked[c * NPROP + id] = dec_score((unsigned)(ki >> 32));
      }
      float areai = (bi.z - bi.x + 1.0f) * (bi.w - bi.y + 1.0f);
      for (int j = i + 1 + tid; j < cnt; j += 256) {
        if (!SUP[j]) {
          float4 bj = BX[j];
          float ix1 = fmaxf(bi.x, bj.x), iy1 = fmaxf(bi.y, bj.y);
          float ix2 = fminf(bi.z, bj.z), iy2 = fminf(bi.w, bj.w);
          float iw = fmaxf(ix2 - ix1 + 1.0f, 0.0f);
          float ih = fmaxf(iy2 - iy1 + 1.0f, 0.0f);
          float inter = iw * ih;
          float areaj = (bj.z - bj.x + 1.0f) * (bj.w - bj.y + 1.0f);
          if (inter / (areai + areaj - inter) > NMS_T) SUP[j] = 1;
        }
      }
    }
    __syncthreads();
  }
}

// =========== Kernel C1: per-1024-chunk top-100 ============================
__global__ __launch_bounds__(256) void kC1(const float* __restrict__ masked,
                                           unsigned long long* __restrict__ blocktop) {
  __shared__ unsigned long long K[1024];
  const int tid = threadIdx.x;
  const int base = blockIdx.x * 1024;
  for (int t = tid; t < 1024; t += 256) {
    unsigned g = base + t;
    K[t] = enc_key(masked[g], g);
  }
  __syncthreads();
  for (int kk = 2; kk <= 1024; kk <<= 1) {
    for (int j = kk >> 1; j > 0; j >>= 1) {
      for (int t = tid; t < 1024; t += 256) {
        int ixj = t ^ j;
        if (ixj > t) {
          unsigned long long a = K[t], b = K[ixj];
          bool up = (t & kk) == 0;
          if (up ? (a < b) : (a > b)) { K[t] = b; K[ixj] = a; }
        }
      }
      __syncthreads();
    }
  }
  for (int t = tid; t < NDET; t += 256)
    blocktop[blockIdx.x * NDET + t] = K[t];
}

// =========== Kernel C2: merge 160x100 -> final 100, write output ==========
__global__ __launch_bounds__(256) void kC2(unsigned long long* __restrict__ blocktop,
                                           const float* __restrict__ boxes_c,
                                           float* __restrict__ out) {
  __shared__ unsigned long long RK[256];
  __shared__ int RP[256];
  const int tid = threadIdx.x;
  const int M = 160 * NDET;
  for (int k = 0; k < NDET; ++k) {
    unsigned long long best = 0ull; int bp = 0;
    for (int t = tid; t < M; t += 256) {
      unsigned long long v = blocktop[t];
      if (v > best) { best = v; bp = t; }
    }
    RK[tid] = best; RP[tid] = bp;
    __syncthreads();
    for (int s = 128; s; s >>= 1) {
      if (tid < s && RK[tid + s] > RK[tid]) { RK[tid] = RK[tid + s]; RP[tid] = RP[tid + s]; }
      __syncthreads();
    }
    if (tid == 0) {
      unsigned long long w = RK[0];
      blocktop[RP[0]] = 0ull;                // mark taken (keys are always > 0)
      unsigned g = ~(unsigned)w;             // flat index = class*2048 + prop
      float sc = dec_score((unsigned)(w >> 32));
      float4 bb = ((const float4*)boxes_c)[g];
      out[k * 4 + 0] = bb.x; out[k * 4 + 1] = bb.y;
      out[k * 4 + 2] = bb.z; out[k * 4 + 3] = bb.w;
      out[4 * NDET + k] = sc;
      out[5 * NDET + k] = (float)(g / NPROP + 1u);
    }
    __syncthreads();
  }
}

extern "C" void kernel_launch(void* const* d_in, const int* in_sizes, int n_in,
                              void* d_out, int out_size, void* d_ws, size_t ws_size,
                              hipStream_t stream) {
  (void)in_sizes; (void)n_in; (void)out_size; (void)ws_size;
  const float* logits = (const float*)d_in[0];   // [2048,81]
  const float* reg    = (const float*)d_in[1];   // [2048,324]
  const float* props  = (const float*)d_in[2];   // [2048,4]
  char* ws = (char*)d_ws;
  float* scores_c = (float*)(ws);                       // 80*2048 f32   (640 KB)
  float* boxes_c  = (float*)(ws + 655360);              // 80*2048*4 f32 (2.5 MB)
  float* masked   = (float*)(ws + 3276800);             // 80*2048 f32   (640 KB)
  unsigned long long* blocktop =
      (unsigned long long*)(ws + 3932160);              // 160*100 u64   (125 KB)
  float* out = (float*)d_out;                           // 400 + 100 + 100

  kA<<<256, 256, 0, stream>>>(logits, reg, props, scores_c, boxes_c);
  kB<<<NFG, 256, 0, stream>>>(scores_c, boxes_c, masked);
  kC1<<<160, 256, 0, stream>>>(masked, blocktop);
  kC2<<<1, 256, 0, stream>>>(blocktop, boxes_c, out);
}